// GNNVectorMultiHeadAttention2_66529043415204
// MI455X (gfx1250) — compile-verified
//
#include <hip/hip_runtime.h>
#include <math.h>

// ---------------------------------------------------------------------------
// CDNA5 (gfx1250) fp32 WMMA implementation of GNN multi-head attention.
// V_WMMA_F32_16X16X4_F32: A (16x4) = float2/lane, B (4x16) = float2/lane,
// C/D (16x16) = 8 floats/lane.  wave32.
// All large matmuls use a double-buffered LDS pipeline fed by
// GLOBAL_LOAD_ASYNC_TO_LDS_B128 (ASYNCcnt) so DMA overlaps the matrix pipe.
// ---------------------------------------------------------------------------

typedef __attribute__((ext_vector_type(2))) float v2f;
typedef __attribute__((ext_vector_type(8))) float v8f;

#define WMMA_F32(a, b, c) \
  __builtin_amdgcn_wmma_f32_16x16x4_f32(false, (a), false, (b), (short)0, (c), false, false)

constexpr int Bsz = 4, LQ = 512, LK = 512, DM = 512, NH = 8, NL = 8, DQ = 64;
constexpr float INV_TEMP = 0.125f;  // 1/sqrt(D_Q)

// Async bulk copy: 16 bytes global -> LDS, tracked by ASYNCcnt (no VGPR data).
__device__ __forceinline__ void async_copy_b128(const float* gsrc, float* lds_dst) {
  unsigned int loff = (unsigned int)(unsigned long long)lds_dst;  // LDS offset = addr[31:0]
  asm volatile("global_load_async_to_lds_b128 %0, %1, off"
               :
               : "v"(loff), "v"(gsrc)
               : "memory");
}
__device__ __forceinline__ void wait_async0() {
  asm volatile("s_wait_asynccnt 0x0" ::: "memory");
}

// ---------------------------------------------------------------------------
// C[M,N] = A[M,K] @ W[N,K]^T (+ bias[N]).  Row-major.
// Block: 256 threads (8 waves) -> 64x128 C macro-tile; wave -> 16x64 strip.
// K pipelined in 32-wide slabs staged to LDS via async b128 copies.
// Requires M%64==0, N%128==0, K%32==0 (true for all uses here).
// ---------------------------------------------------------------------------
#define KT 32
__global__ void __launch_bounds__(256) gemm_xt_wmma(
    const float* __restrict__ A, const float* __restrict__ W,
    const float* __restrict__ bias, float* __restrict__ C,
    int M, int N, int K, int lda, int ldw, int ldc) {
  __shared__ float As[2][64][KT];   //  8 KB per buffer
  __shared__ float Ws[2][128][KT];  // 16 KB per buffer

  const int tilesN = N >> 7;
  const int bm = blockIdx.x / tilesN, bn = blockIdx.x - bm * tilesN;
  const int m0 = bm << 6, n0 = bn << 7;
  const int t = threadIdx.x;
  const int lane = t & 31;
  const int wave = t >> 5;
  const int wm = wave >> 1, wn = wave & 1;  // 4x2 wave grid
  const int mr = lane & 15;
  const int kh = (lane >> 4) << 1;  // lanes 16..31 hold K+2,K+3

  // Stage one K-slab (A: 64x32, W: 128x32): 256 threads x (2+4) b128 copies.
  auto stage = [&](int bs, int kt) {
#pragma unroll
    for (int i = 0; i < 2; ++i) {
      int v = t + (i << 8);
      int row = v >> 3;  // 8 x float4 per 32-float row
      int c4 = (v & 7) << 2;
      async_copy_b128(A + (size_t)(m0 + row) * lda + kt + c4, &As[bs][row][c4]);
    }
#pragma unroll
    for (int i = 0; i < 4; ++i) {
      int v = t + (i << 8);
      int row = v >> 3;
      int c4 = (v & 7) << 2;
      async_copy_b128(W + (size_t)(n0 + row) * ldw + kt + c4, &Ws[bs][row][c4]);
    }
  };

  stage(0, 0);
  v8f acc[4] = {v8f{}, v8f{}, v8f{}, v8f{}};
  int buf = 0;
  for (int kt = 0; kt < K; kt += KT) {
    wait_async0();    // my async copies into `buf` have landed
    __syncthreads();  // everyone's have; `buf` fully populated
    if (kt + KT < K) stage(buf ^ 1, kt + KT);  // overlap next slab with compute
    const float* ap = &As[buf][(wm << 4) + mr][kh];
    const float* wp = &Ws[buf][(wn << 6) + mr][kh];
#pragma unroll
    for (int kk = 0; kk < KT; kk += 4) {
      v2f a = *(const v2f*)(ap + kk);
#pragma unroll
      for (int j = 0; j < 4; ++j) {
        v2f bf = *(const v2f*)(wp + j * 16 * KT + kk);
        acc[j] = WMMA_F32(a, bf, acc[j]);
      }
    }
    __syncthreads();  // all waves done reading `buf` before it is re-staged
    buf ^= 1;
  }

  const int rh = (lane >> 4) << 3;
#pragma unroll
  for (int j = 0; j < 4; ++j) {
    int col = n0 + (wn << 6) + (j << 4) + mr;
    float bv = bias ? bias[col] : 0.0f;
#pragma unroll
    for (int r = 0; r < 8; ++r) {
      C[(size_t)(m0 + (wm << 4) + rh + r) * ldc + col] = acc[j][r] + bv;
    }
  }
}

// ---------------------------------------------------------------------------
// qb[h,b,q,l] = dot(Qp[b,q, h*64 : h*64+64], b_ks[l,:])
// ---------------------------------------------------------------------------
__global__ void qb_kernel(const float* __restrict__ Qp, const float* __restrict__ b_ks,
                          float* __restrict__ qb) {
  int idx = blockIdx.x * blockDim.x + threadIdx.x;  // NH*B*LQ*NL
  if (idx >= NH * Bsz * LQ * NL) return;
  int l = idx & 7;
  int t = idx >> 3;  // hb*LQ + q
  int q = t & (LQ - 1);
  int hb = t >> 9;
  int h = hb >> 2, b = hb & 3;
  const float* qrow = Qp + (size_t)(b * LQ + q) * DM + h * DQ;
  const float* krow = b_ks + l * DQ;
  float s = 0.f;
  for (int d = 0; d < DQ; ++d) s += qrow[d] * krow[d];
  qb[idx] = s;
}

// ---------------------------------------------------------------------------
// rel[b,q,k] = 1 if (sum_l edge[b,q,k,l] + 1 - pad[b,q,k]) == 0 else 0
// ---------------------------------------------------------------------------
__global__ void relmask_kernel(const float* __restrict__ edge, const float* __restrict__ pad,
                               float* __restrict__ rel) {
  int idx = blockIdx.x * blockDim.x + threadIdx.x;  // B*LQ*LK
  if (idx >= Bsz * LQ * LK) return;
  const float4* e4 = (const float4*)(edge + (size_t)idx * NL);
  float4 a = e4[0], c = e4[1];
  float s = a.x + a.y + a.z + a.w + c.x + c.y + c.z + c.w;
  s += 1.0f - pad[idx];
  rel[idx] = (s == 0.0f) ? 1.0f : 0.0f;
}

// ---------------------------------------------------------------------------
// S[h,b,q,k] = (Q·K^T + sum_l edge[b,q,k,l]*qb[h,b,q,l]) / temp
// One wave -> 16x64 score strip; K-dim = 64 (16 WMMA steps).
// Edge vectors fetched as 2 x b128.
// ---------------------------------------------------------------------------
__global__ void scores_wmma(const float* __restrict__ Qp, const float* __restrict__ Kp,
                            const float* __restrict__ qb, const float* __restrict__ edge,
                            float* __restrict__ Sout) {
  int wave = (blockIdx.x * blockDim.x + threadIdx.x) >> 5;
  int lane = threadIdx.x & 31;
  int hb = wave >> 8;  // 256 tiles per (h,b): 32 q-tiles x 8 k-strips
  int rem = wave & 255;
  int qt = rem >> 3, kt = rem & 7;
  int h = hb >> 2, b = hb & 3;
  int q0 = qt << 4, k0 = kt << 6;
  int mr = lane & 15;
  int kh = (lane >> 4) << 1;
  const float* ap = Qp + (size_t)(b * LQ + q0 + mr) * DM + h * DQ + kh;
  const float* bp = Kp + (size_t)(b * LK + k0 + mr) * DM + h * DQ + kh;
  v8f acc[4] = {v8f{}, v8f{}, v8f{}, v8f{}};
#pragma unroll
  for (int kk = 0; kk < DQ; kk += 4) {
    v2f a = *(const v2f*)(ap + kk);
#pragma unroll
    for (int j = 0; j < 4; ++j) {
      v2f bf = *(const v2f*)(bp + (size_t)(j << 4) * DM + kk);
      acc[j] = WMMA_F32(a, bf, acc[j]);
    }
  }
  int rh = (lane >> 4) << 3;
#pragma unroll
  for (int r = 0; r < 8; ++r) {
    int q = q0 + rh + r;
    const float* qv = qb + ((size_t)hb * LQ + q) * NL;
    float qv8[NL];
#pragma unroll
    for (int l = 0; l < NL; ++l) qv8[l] = qv[l];
#pragma unroll
    for (int j = 0; j < 4; ++j) {
      int k = k0 + (j << 4) + mr;
      const float4* e4 = (const float4*)(edge + ((size_t)(b * LQ + q) * LK + k) * NL);
      float4 ea = e4[0], eb = e4[1];
      float s = acc[j][r];
      s += ea.x * qv8[0] + ea.y * qv8[1] + ea.z * qv8[2] + ea.w * qv8[3];
      s += eb.x * qv8[4] + eb.y * qv8[5] + eb.z * qv8[6] + eb.w * qv8[7];
      Sout[((size_t)hb * LQ + q) * LK + k] = s * INV_TEMP;
    }
  }
}

// ---------------------------------------------------------------------------
// Masked softmax over k (512).  One 256-thread block per (h,b,q) row.
// Fully-masked rows produce zeros (matches reference where(rm,0,softmax)).
// ---------------------------------------------------------------------------
__global__ void softmax_kernel(const float* __restrict__ S, const float* __restrict__ rel,
                               float* __restrict__ ATT) {
  int row = blockIdx.x;  // hb*LQ + q
  int q = row & (LQ - 1);
  int hb = row >> 9;
  int b = hb & 3;
  int t = threadIdx.x;
  const float* srow = S + (size_t)row * LK;
  const float* rrow = rel + (size_t)(b * LQ + q) * LK;
  float x0 = srow[t], x1 = srow[t + 256];
  float r0 = rrow[t], r1 = rrow[t + 256];
  float y0 = (r0 != 0.f) ? -INFINITY : x0;
  float y1 = (r1 != 0.f) ? -INFINITY : x1;
  __shared__ float red[256];
  red[t] = fmaxf(y0, y1);
  __syncthreads();
  for (int s = 128; s > 0; s >>= 1) {
    if (t < s) red[t] = fmaxf(red[t], red[t + s]);
    __syncthreads();
  }
  float m = red[0];
  __syncthreads();
  float e0 = (r0 != 0.f) ? 0.f : __expf(y0 - m);
  float e1 = (r1 != 0.f) ? 0.f : __expf(y1 - m);
  red[t] = e0 + e1;
  __syncthreads();
  for (int s = 128; s > 0; s >>= 1) {
    if (t < s) red[t] += red[t + s];
    __syncthreads();
  }
  float sum = red[0];
  float inv = (sum > 0.f) ? 1.0f / sum : 0.0f;
  ATT[(size_t)row * LK + t] = e0 * inv;
  ATT[(size_t)row * LK + t + 256] = e1 * inv;
}

// ---------------------------------------------------------------------------
// QV[h,b,q,d] = sum_k ATT[h,b,q,k] * Vp[b,k, h*64+d].
// Block (256 thr, 8 waves) per (h,b, 128-q-row group); all waves share the
// V K-slab (32x64) staged in double-buffered LDS via async b128 copies.
// Wave -> 16x64 tile (full head dim); A rows stream from L2 as float2.
// ---------------------------------------------------------------------------
#define KVT 32
__global__ void __launch_bounds__(256) qv_wmma(const float* __restrict__ ATT,
                                               const float* __restrict__ Vp,
                                               float* __restrict__ QV) {
  __shared__ float Vs[2][KVT][DQ];  // 8 KB per buffer
  const int hb = blockIdx.x >> 2;   // 32 (h,b) pairs
  const int qg = blockIdx.x & 3;    // 4 groups of 8 q-tiles
  const int h = hb >> 2, b = hb & 3;
  const int t = threadIdx.x;
  const int lane = t & 31;
  const int wave = t >> 5;
  const int q0 = ((qg << 3) + wave) << 4;
  const int mr = lane & 15;
  const int kh = (lane >> 4) << 1;

  const float* vbase = Vp + (size_t)(b * LK) * DM + h * DQ;
  // Stage V rows kt..kt+31 (64 wide): 512 float4 -> 2 copies per thread.
  auto stage = [&](int bs, int kt) {
#pragma unroll
    for (int i = 0; i < 2; ++i) {
      int v = t + (i << 8);
      int row = v >> 4;  // 16 x float4 per 64-float row
      int c4 = (v & 15) << 2;
      async_copy_b128(vbase + (size_t)(kt + row) * DM + c4, &Vs[bs][row][c4]);
    }
  };

  const float* ap = ATT + ((size_t)hb * LQ + q0 + mr) * LK + kh;
  stage(0, 0);
  v8f acc[4] = {v8f{}, v8f{}, v8f{}, v8f{}};
  int buf = 0;
  for (int kt = 0; kt < LK; kt += KVT) {
    wait_async0();
    __syncthreads();
    if (kt + KVT < LK) stage(buf ^ 1, kt + KVT);
#pragma unroll
    for (int kk = 0; kk < KVT; kk += 4) {
      v2f a = *(const v2f*)(ap + kt + kk);
      const float* vr = &Vs[buf][kk + kh][0];
#pragma unroll
      for (int j = 0; j < 4; ++j) {
        int n = (j << 4) + mr;
        v2f bf = {vr[n], vr[DQ + n]};  // V[k][n], V[k+1][n]
        acc[j] = WMMA_F32(a, bf, acc[j]);
      }
    }
    __syncthreads();
    buf ^= 1;
  }
  int rh = (lane >> 4) << 3;
#pragma unroll
  for (int j = 0; j < 4; ++j) {
    int d = (j << 4) + mr;
#pragma unroll
    for (int r = 0; r < 8; ++r) {
      int q = q0 + rh + r;
      QV[((size_t)hb * LQ + q) * DQ + d] = acc[j][r];
    }
  }
}

// ---------------------------------------------------------------------------
// aw[l] = sum_k ATT[h,b,q,k]*edge[b,q,k,l];  out_pre[b,q,h*64+d] =
//   QV[h,b,q,d] + sum_l aw[l]*b_vs[l,d].  64-thread block per (h,b,q).
// ---------------------------------------------------------------------------
__global__ void combine_kernel(const float* __restrict__ QV, const float* __restrict__ ATT,
                               const float* __restrict__ edge, const float* __restrict__ b_vs,
                               float* __restrict__ OutPre) {
  int row = blockIdx.x;  // hb*LQ + q
  int q = row & (LQ - 1);
  int hb = row >> 9;
  int h = hb >> 2, b = hb & 3;
  int t = threadIdx.x;  // 0..63
  __shared__ float part[64];
  __shared__ float aw[NL];
  int l = t & 7, g = t >> 3;
  const float* arow = ATT + (size_t)row * LK;
  const float* erow = edge + (size_t)(b * LQ + q) * LK * NL;
  float s = 0.f;
  for (int k = g; k < LK; k += 8) s += arow[k] * erow[(size_t)k * NL + l];
  part[t] = s;
  __syncthreads();
  if (t < NL) {
    float a = 0.f;
#pragma unroll
    for (int gg = 0; gg < 8; ++gg) a += part[t + 8 * gg];
    aw[t] = a;
  }
  __syncthreads();
  int d = t;
  float o = QV[(size_t)row * DQ + d];
#pragma unroll
  for (int ll = 0; ll < NL; ++ll) o += aw[ll] * b_vs[ll * DQ + d];
  OutPre[(size_t)(b * LQ + q) * DM + h * DQ + d] = o;
}

// ---------------------------------------------------------------------------
extern "C" void kernel_launch(void* const* d_in, const int* in_sizes, int n_in,
                              void* d_out, int out_size, void* d_ws, size_t ws_size,
                              hipStream_t stream) {
  const float* q = (const float*)d_in[0];
  const float* k = (const float*)d_in[1];
  const float* edge = (const float*)d_in[2];
  const float* pad = (const float*)d_in[3];
  const float* w_q_w = (const float*)d_in[4];
  const float* w_q_b = (const float*)d_in[5];
  const float* w_k = (const float*)d_in[6];
  const float* w_v = (const float*)d_in[7];
  const float* w_h_w = (const float*)d_in[8];
  const float* w_h_b = (const float*)d_in[9];
  const float* b_ks = (const float*)d_in[10];
  const float* b_vs = (const float*)d_in[11];

  float* out = (float*)d_out;                 // (B, LQ, DM)
  float* Satt = out + (size_t)Bsz * LQ * DM;  // (NH*B, LQ, LK) pre-mask weights

  // Workspace (floats).  QV aliases Kp, OutPre aliases Qp (dead by then).
  float* ws = (float*)d_ws;
  const size_t PROJ = (size_t)Bsz * LQ * DM;  // 1048576
  float* Qp = ws;
  float* Kp = Qp + PROJ;
  float* Vp = Kp + PROJ;
  float* qb = Vp + PROJ;                         // NH*B*LQ*NL = 131072
  float* rel = qb + (size_t)NH * Bsz * LQ * NL;  // B*LQ*LK   = 1048576
  float* ATT = rel + (size_t)Bsz * LQ * LK;      // NH*B*LQ*LK = 8388608
  float* QV = Kp;                                // alias (Kp dead after scores)
  float* OutPre = Qp;                            // alias (Qp dead after scores/qb)

  const int M = Bsz * LQ;  // 2048
  dim3 blk256(256), blk64(64);
  int gemm_blocks = (M / 64) * (DM / 128);  // 128 blocks, 64x128 macro-tiles

  // 1-3: Q/K/V projections (C = A @ W^T [+ bias]) — async-LDS WMMA pipeline
  gemm_xt_wmma<<<gemm_blocks, blk256, 0, stream>>>(q, w_q_w, w_q_b, Qp, M, DM, DM, DM, DM, DM);
  gemm_xt_wmma<<<gemm_blocks, blk256, 0, stream>>>(k, w_k, nullptr, Kp, M, DM, DM, DM, DM, DM);
  gemm_xt_wmma<<<gemm_blocks, blk256, 0, stream>>>(k, w_v, nullptr, Vp, M, DM, DM, DM, DM, DM);

  // 4: qb[h,b,q,l]
  qb_kernel<<<(NH * Bsz * LQ * NL) / 256, blk256, 0, stream>>>(Qp, b_ks, qb);

  // 5: relation mask
  relmask_kernel<<<(Bsz * LQ * LK) / 256, blk256, 0, stream>>>(edge, pad, rel);

  // 6: scores -> d_out attention region (8192 waves -> 1024 blocks)
  scores_wmma<<<1024, blk256, 0, stream>>>(Qp, Kp, qb, edge, Satt);

  // 7: masked softmax (one block per row)
  softmax_kernel<<<NH * Bsz * LQ, blk256, 0, stream>>>(Satt, rel, ATT);

  // 8: attn @ V — async-LDS pipelined (128 blocks x 8 waves)
  qv_wmma<<<128, blk256, 0, stream>>>(ATT, Vp, QV);

  // 9: add relative-value term, pack heads back to (B, LQ, DM)
  combine_kernel<<<NH * Bsz * LQ, blk64, 0, stream>>>(QV, ATT, edge, b_vs, OutPre);

  // 10: output projection into d_out
  gemm_xt_wmma<<<gemm_blocks, blk256, 0, stream>>>(OutPre, w_h_w, w_h_b, out, M, DM, DM, DM, DM, DM);
}